// CenterLoss_4252017623275
// MI455X (gfx1250) — compile-verified
//
#include <hip/hip_runtime.h>

typedef __attribute__((ext_vector_type(2))) float v2f;
typedef __attribute__((ext_vector_type(8))) float v8f;

#define WAVES_PER_BLOCK 4
#define NUM_TILES 128           // 2048 samples / 16 per tile
#define FEAT 64

// Each wave computes one 16-sample tile:
//   delta = x_tile - gathered_center_tile        (16 x 64, f32)
//   D     = delta @ delta^T  via 16 chained v_wmma_f32_16x16x4_f32
//   dist  = diag(D), clipped, wave-summed -> one partial per tile
__global__ __launch_bounds__(128) void centerloss_wmma_kernel(
    const float* __restrict__ x,          // (2048, 64)
    const long long* __restrict__ labels, // (2048,) int64
    const float* __restrict__ centers,    // (100000, 64)
    float* __restrict__ partial)          // (128,) per-tile sums
{
    const int lane = threadIdx.x & 31;
    const int wave = threadIdx.x >> 5;
    const int tile = blockIdx.x * WAVES_PER_BLOCK + wave;   // 0..127
    const int row  = lane & 15;                             // M index in tile
    const int half = lane >> 4;                             // 0 or 1 (K phase)
    const int s    = tile * 16 + row;                       // sample index

    const long long lbl = labels[s];
    const float* __restrict__ xr = x + (size_t)s * FEAT;
    const float* __restrict__ cr = centers + (size_t)lbl * FEAT;

    // A-fragment for 16x4 f32 tile j: lane holds Delta[row][4j+2h], Delta[row][4j+2h+1].
    // B-fragment (Delta^T, 4x16) for the same lane is the identical pair -> reuse.
    v2f frag[16];
#pragma unroll
    for (int j = 0; j < 16; ++j) {
        const int col = 4 * j + 2 * half;
        float2 xa = *(const float2*)(xr + col);
        float2 ca = *(const float2*)(cr + col);
        v2f a;
        a.x = xa.x - ca.x;
        a.y = xa.y - ca.y;
        frag[j] = a;
    }

    v8f acc = {};
#pragma unroll
    for (int j = 0; j < 16; ++j) {
        // (neg_a, A, neg_b, B, c_mod, C, reuse_a, reuse_b)
        acc = __builtin_amdgcn_wmma_f32_16x16x4_f32(
            false, frag[j], false, frag[j], (short)0, acc, false, false);
    }

    // Diagonal of 16x16 f32 D: lanes 0-7 hold diag rows 0-7 in acc[lane];
    // lanes 24-31 hold diag rows 8-15 in acc[lane-24]. Other lanes contribute 0.
    const bool valid = (lane < 8) || (lane >= 24);
    const int  sel   = (lane < 8) ? lane : (lane - 24);
    float dv = acc[0];
#pragma unroll
    for (int i = 1; i < 8; ++i)
        if (sel == i) dv = acc[i];

    float dist = valid ? fminf(fmaxf(dv, 1e-12f), 1e12f) : 0.0f;

    // wave32 all-lane sum
#pragma unroll
    for (int off = 16; off >= 1; off >>= 1)
        dist += __shfl_xor(dist, off, 32);

    if (lane == 0)
        partial[tile] = dist;
}

__global__ __launch_bounds__(128) void centerloss_reduce_kernel(
    const float* __restrict__ partial, float* __restrict__ out)
{
    __shared__ float smem[NUM_TILES];
    const int t = threadIdx.x;
    smem[t] = partial[t];
    __syncthreads();
#pragma unroll
    for (int off = NUM_TILES / 2; off > 0; off >>= 1) {
        if (t < off) smem[t] += smem[t + off];
        __syncthreads();
    }
    if (t == 0)
        out[0] = smem[0] * (1.0f / 2048.0f);
}

extern "C" void kernel_launch(void* const* d_in, const int* in_sizes, int n_in,
                              void* d_out, int out_size, void* d_ws, size_t ws_size,
                              hipStream_t stream) {
    const float*     x       = (const float*)d_in[0];
    const long long* labels  = (const long long*)d_in[1];
    const float*     centers = (const float*)d_in[2];
    float* out = (float*)d_out;
    float* ws  = (float*)d_ws;   // 128 floats of scratch, rewritten every call

    centerloss_wmma_kernel<<<NUM_TILES / WAVES_PER_BLOCK, 32 * WAVES_PER_BLOCK, 0, stream>>>(
        x, labels, centers, ws);
    centerloss_reduce_kernel<<<1, NUM_TILES, 0, stream>>>(ws, out);
}